// GDN_12524124635910
// MI455X (gfx1250) — compile-verified
//
#include <hip/hip_runtime.h>
#include <hip/hip_bf16.h>

#define N_NODES 1024
#define FDIM 64
#define DDIM 64
#define TOPK 20
#define BATCH 64
#define BN_CNT 65536.0f   // B * N
#define EPS 1e-5f
#define NEG_SLOPE 0.2f

typedef __attribute__((ext_vector_type(2))) float v2f;
typedef __attribute__((ext_vector_type(8))) float v8f;

// ---------- workspace layout (floats) ----------
#define WN_OFF   0            // 1024*64 normalized embedding
#define EI_OFF   65536        // 1024  : embedding @ att_em_i
#define EJ_OFF   66560        // 1024  : embedding @ att_em_j
#define XL_OFF   67584        // 65536*64 : data @ lin_w
#define ST_OFF   4261888      // 256 : bn1sum[64] bn1sq[64] bn2sum[64] bn2sq[64]
#define TOPI_OFF 4262144      // 1024*20 ints

#define LDA 66                // padded A row stride (floats) -> conflict-free ds reads

__device__ __forceinline__ float wsum32(float v) {
#pragma unroll
    for (int o = 16; o > 0; o >>= 1) v += __shfl_xor(v, o, 32);
    return v;
}

// ---------------- K_zero: clear stat accumulators ----------------
__global__ void k_zero_stats(float* __restrict__ stats) {
    if (threadIdx.x < 256) stats[threadIdx.x] = 0.0f;
}

// ---------------- K0: normalize embedding, e_i / e_j dots --------
__global__ void k_prep(const float* __restrict__ emb,
                       const float* __restrict__ att_em_i,
                       const float* __restrict__ att_em_j,
                       float* __restrict__ wn,
                       float* __restrict__ e_i,
                       float* __restrict__ e_j) {
    const int wave = threadIdx.x >> 5;
    const int lane = threadIdx.x & 31;
    const int r = blockIdx.x * 8 + wave;              // one wave per row
    const float2 x = *(const float2*)(emb + r * DDIM + 2 * lane);
    float n2 = wsum32(x.x * x.x + x.y * x.y);
    float inv = rsqrtf(fmaxf(n2, 1e-20f));
    float2 w; w.x = x.x * inv; w.y = x.y * inv;
    *(float2*)(wn + r * DDIM + 2 * lane) = w;
    const float2 ai = *(const float2*)(att_em_i + 2 * lane);
    const float2 aj = *(const float2*)(att_em_j + 2 * lane);
    float si = wsum32(x.x * ai.x + x.y * ai.y);
    float sj = wsum32(x.x * aj.x + x.y * aj.y);
    if (lane == 0) { e_i[r] = si; e_j[r] = sj; }
}

// ---------------- K1: cosine row + top-20 selection --------------
__global__ void k_topk(const float* __restrict__ wn, int* __restrict__ topi) {
    __shared__ float srow[DDIM];
    __shared__ float sc[N_NODES];
    __shared__ float rv[256];
    __shared__ int   ri[256];
    const int r = blockIdx.x;
    const int t = threadIdx.x;
    if (t < DDIM) srow[t] = wn[r * DDIM + t];
    __syncthreads();
    for (int c = t; c < N_NODES; c += 256) {
        const float4* w4 = (const float4*)(wn + c * DDIM);
        float d = 0.0f;
#pragma unroll
        for (int q = 0; q < 16; ++q) {
            float4 v = w4[q];
            d += v.x * srow[4 * q] + v.y * srow[4 * q + 1] +
                 v.z * srow[4 * q + 2] + v.w * srow[4 * q + 3];
        }
        sc[c] = d;
    }
    __syncthreads();
    for (int k = 0; k < TOPK; ++k) {
        float best = -3.0e38f; int bi = N_NODES;
        for (int c = t; c < N_NODES; c += 256) {
            float v = sc[c];
            if (v > best || (v == best && c < bi)) { best = v; bi = c; }
        }
        rv[t] = best; ri[t] = bi;
        __syncthreads();
        for (int s = 128; s > 0; s >>= 1) {
            if (t < s) {
                float v = rv[t + s]; int j = ri[t + s];
                if (v > rv[t] || (v == rv[t] && j < ri[t])) { rv[t] = v; ri[t] = j; }
            }
            __syncthreads();
        }
        if (t == 0) { topi[r * TOPK + k] = ri[0]; sc[ri[0]] = -3.0e38f; }
        __syncthreads();
    }
}

// -------- K2: xl = data @ lin_w, LDS-staged V_WMMA_F32_16X16X4 ---
// Block: 256 threads (8 waves), 32 rows x 64 cols of output.
// A tile (32x64) loaded coalesced (b128) into padded LDS (stride 66);
// B (64x64 lin_w) loaded coalesced into LDS (stride 64, conflict-free
// for the B operand since lanes address consecutive columns).
__global__ void k_gemm(const float* __restrict__ data,
                       const float* __restrict__ linw,
                       float* __restrict__ xl) {
    __shared__ float ldsA[32 * LDA];
    __shared__ float ldsB[64 * 64];
    const int t = threadIdx.x;
    const int rowBase = blockIdx.x * 32;

    // A tile: 32 rows * 16 float4 = 512 float4s; 2 per thread, coalesced
#pragma unroll
    for (int q = 0; q < 2; ++q) {
        const int f4 = t + q * 256;          // 0..511
        const int r  = f4 >> 4;              // row in tile
        const int c4 = f4 & 15;              // float4 within row
        const float4 v = *(const float4*)(data + (size_t)(rowBase + r) * FDIM + c4 * 4);
        const int la = r * LDA + c4 * 4;
        ldsA[la + 0] = v.x; ldsA[la + 1] = v.y;
        ldsA[la + 2] = v.z; ldsA[la + 3] = v.w;
    }
    // B: 64x64 = 1024 float4s; 4 per thread, coalesced
#pragma unroll
    for (int q = 0; q < 4; ++q) {
        const int f4 = t + q * 256;
        *(float4*)(ldsB + f4 * 4) = *(const float4*)(linw + f4 * 4);
    }
    __syncthreads();

    const int wave = t >> 5;
    const int lane = t & 31;
    const int rtl = (wave >> 2) * 16;        // local row tile
    const int colTile = (wave & 3) * 16;
    const int lh = lane >> 4;                // half-wave K selector
    const int l  = lane & 15;
    v8f acc = {};
#pragma unroll
    for (int k0 = 0; k0 < FDIM; k0 += 4) {
        const int kk = k0 + 2 * lh;
        v2f a, b;
        a.x = ldsA[(rtl + l) * LDA + kk];
        a.y = ldsA[(rtl + l) * LDA + kk + 1];
        b.x = ldsB[kk * 64 + colTile + l];
        b.y = ldsB[(kk + 1) * 64 + colTile + l];
        acc = __builtin_amdgcn_wmma_f32_16x16x4_f32(
            false, a, false, b, (short)0, acc, false, false);
    }
#pragma unroll
    for (int v = 0; v < 8; ++v) {
        const int r = rowBase + rtl + v + 8 * lh;
        xl[(size_t)r * DDIM + colTile + l] = acc[v];
    }
}

// ---------------- K3: fused attention + aggregation + BN1 stats --
__global__ void k_attn(const float* __restrict__ xl,
                       const int* __restrict__ topi,
                       const float* __restrict__ e_i,
                       const float* __restrict__ e_j,
                       const float* __restrict__ att_i,
                       const float* __restrict__ att_j,
                       const float* __restrict__ gnn_bias,
                       float* __restrict__ gnn,          // d_out "out" region
                       float* __restrict__ stats) {
    __shared__ float ssum[DDIM];
    __shared__ float ssq[DDIM];
    const int wave = threadIdx.x >> 5;
    const int lane = threadIdx.x & 31;
    if (threadIdx.x < DDIM) { ssum[threadIdx.x] = 0.0f; ssq[threadIdx.x] = 0.0f; }
    __syncthreads();

    const int idx = blockIdx.x * 8 + wave;            // flattened (b,n)
    const int b = idx >> 10;
    const int n = idx & (N_NODES - 1);
    const int c0 = 2 * lane;

    const float2 ati = *(const float2*)(att_i + c0);
    const float2 atj = *(const float2*)(att_j + c0);
    const float2 own = *(const float2*)(xl + (size_t)idx * DDIM + c0);
    const float base = wsum32(own.x * ati.x + own.y * ati.y) + e_i[n];

    float2 xj[TOPK];
    float  sc[TOPK];
    float  m = -3.0e38f;
#pragma unroll
    for (int k = 0; k < TOPK; ++k) {
        const int nbr = topi[n * TOPK + k];
        const float2 xv = *(const float2*)(xl + (size_t)(b * N_NODES + nbr) * DDIM + c0);
        xj[k] = xv;
        float s = wsum32(xv.x * atj.x + xv.y * atj.y) + e_j[nbr] + base;
        s = (s > 0.0f) ? s : NEG_SLOPE * s;           // leaky relu
        sc[k] = s;
        m = fmaxf(m, s);
    }
    float den = 0.0f;
#pragma unroll
    for (int k = 0; k < TOPK; ++k) { sc[k] = __expf(sc[k] - m); den += sc[k]; }
    const float rden = 1.0f / den;
    const float2 gb = *(const float2*)(gnn_bias + c0);
    float2 acc; acc.x = gb.x; acc.y = gb.y;
#pragma unroll
    for (int k = 0; k < TOPK; ++k) {
        const float a = sc[k] * rden;
        acc.x += a * xj[k].x;
        acc.y += a * xj[k].y;
    }
    *(float2*)(gnn + (size_t)idx * DDIM + c0) = acc;

    atomicAdd(&ssum[c0], acc.x);        atomicAdd(&ssum[c0 + 1], acc.y);
    atomicAdd(&ssq[c0], acc.x * acc.x); atomicAdd(&ssq[c0 + 1], acc.y * acc.y);
    __syncthreads();
    if (threadIdx.x < DDIM) {
        atomicAdd(&stats[threadIdx.x], ssum[threadIdx.x]);          // bn1 sum
        atomicAdd(&stats[64 + threadIdx.x], ssq[threadIdx.x]);      // bn1 sumsq
    }
}

// ---------------- K5: apply BN1, multiply embedding, BN2 stats ---
__global__ void k_bn1mul(float* __restrict__ buf,       // in/out (gnn -> pre-bn2)
                         const float* __restrict__ emb,
                         const float* __restrict__ g1,
                         const float* __restrict__ b1,
                         float* __restrict__ stats) {
    __shared__ float ssum[256];
    __shared__ float ssq[256];
    const int t = threadIdx.x;
    const size_t base = (size_t)blockIdx.x * 1024;
    const int c = t & 63;
    const float mu = stats[c] / BN_CNT;
    const float var = stats[64 + c] / BN_CNT - mu * mu;
    const float rs = rsqrtf(var + EPS);
    const float gg = g1[c], bb = b1[c];
    float ls = 0.0f, lq = 0.0f;
#pragma unroll
    for (int j = 0; j < 4; ++j) {
        const size_t i = base + (size_t)j * 256 + t;
        const int n = (int)((i >> 6) & (N_NODES - 1));
        float v = (gg * (buf[i] - mu) * rs + bb) * emb[n * DDIM + c];
        buf[i] = v;
        ls += v; lq += v * v;
    }
    ssum[t] = ls; ssq[t] = lq;
    __syncthreads();
    if (t < 64) {
        float s = ssum[t] + ssum[t + 64] + ssum[t + 128] + ssum[t + 192];
        float q = ssq[t] + ssq[t + 64] + ssq[t + 128] + ssq[t + 192];
        atomicAdd(&stats[128 + t], s);
        atomicAdd(&stats[192 + t], q);
    }
}

// ---------------- K6: BN2 + relu + output projection -------------
__global__ void k_final(float* __restrict__ buf,        // in/out -> final "out"
                        const float* __restrict__ g2,
                        const float* __restrict__ b2,
                        const float* __restrict__ out_w,
                        const float* __restrict__ out_b,
                        const float* __restrict__ stats,
                        float* __restrict__ pred) {
    const int wave = threadIdx.x >> 5;
    const int lane = threadIdx.x & 31;
    const int idx = blockIdx.x * 8 + wave;
    const int c0 = 2 * lane;
    float mu0 = stats[128 + c0] / BN_CNT;
    float mu1 = stats[128 + c0 + 1] / BN_CNT;
    float rs0 = rsqrtf(stats[192 + c0] / BN_CNT - mu0 * mu0 + EPS);
    float rs1 = rsqrtf(stats[192 + c0 + 1] / BN_CNT - mu1 * mu1 + EPS);
    float2 v = *(const float2*)(buf + (size_t)idx * DDIM + c0);
    float2 o;
    o.x = fmaxf(g2[c0] * (v.x - mu0) * rs0 + b2[c0], 0.0f);
    o.y = fmaxf(g2[c0 + 1] * (v.y - mu1) * rs1 + b2[c0 + 1], 0.0f);
    *(float2*)(buf + (size_t)idx * DDIM + c0) = o;
    float p = wsum32(o.x * out_w[c0] + o.y * out_w[c0 + 1]);
    if (lane == 0) pred[idx] = p + out_b[0];
}

extern "C" void kernel_launch(void* const* d_in, const int* in_sizes, int n_in,
                              void* d_out, int out_size, void* d_ws, size_t ws_size,
                              hipStream_t stream) {
    const float* data     = (const float*)d_in[0];
    const float* emb      = (const float*)d_in[2];
    const float* lin_w    = (const float*)d_in[3];
    const float* att_i    = (const float*)d_in[4];
    const float* att_j    = (const float*)d_in[5];
    const float* att_em_i = (const float*)d_in[6];
    const float* att_em_j = (const float*)d_in[7];
    const float* gnn_bias = (const float*)d_in[8];
    const float* bn1_g    = (const float*)d_in[9];
    const float* bn1_b    = (const float*)d_in[10];
    const float* bn2_g    = (const float*)d_in[11];
    const float* bn2_b    = (const float*)d_in[12];
    const float* out_w    = (const float*)d_in[13];
    const float* out_b    = (const float*)d_in[14];

    float* ws   = (float*)d_ws;
    float* wn   = ws + WN_OFF;
    float* e_i  = ws + EI_OFF;
    float* e_j  = ws + EJ_OFF;
    float* xl   = ws + XL_OFF;
    float* stats= ws + ST_OFF;
    int*   topi = (int*)(ws + TOPI_OFF);

    float* pred = (float*)d_out;                 // B*N
    float* obuf = pred + BATCH * N_NODES;        // B*N*D (staged in place)

    k_zero_stats<<<1, 256, 0, stream>>>(stats);
    k_prep<<<128, 256, 0, stream>>>(emb, att_em_i, att_em_j, wn, e_i, e_j);
    k_topk<<<N_NODES, 256, 0, stream>>>(wn, topi);
    k_gemm<<<2048, 256, 0, stream>>>(data, lin_w, xl);
    k_attn<<<8192, 256, 0, stream>>>(xl, topi, e_i, e_j, att_i, att_j,
                                     gnn_bias, obuf, stats);
    k_bn1mul<<<4096, 256, 0, stream>>>(obuf, emb, bn1_g, bn1_b, stats);
    k_final<<<8192, 256, 0, stream>>>(obuf, bn2_g, bn2_b, out_w, out_b, stats, pred);
}